// CNN_85014582657481
// MI455X (gfx1250) — compile-verified
//
#include <hip/hip_runtime.h>
#include <math.h>

typedef float v2f __attribute__((ext_vector_type(2)));
typedef float v8f __attribute__((ext_vector_type(8)));

#define SEQ 123
#define SP  128      // padded sequence
#define DD  16
#define NL  5
#define FFD 64

// ---- d_out flat offsets (return-tuple order) ----
#define O_OUT    0        // (1,2)
#define O_AEA    2        // (123,123)
#define O_WAVB   15131    // wavA_before (123,16)
#define O_WAVA   17099    // wavA_after  (123,16)
#define O_AEFC   19067    // ae_fcA (123,64)
#define O_EAA    26939
#define O_AEB    42068
#define O_EAB    57197
#define O_AEFCL  72326    // ae_fc_lastA (123,16)

// ---- workspace float offsets ----
#define W_POS   0
#define W_EEGT  2048
#define W_WA    4096
#define W_WB    6144
#define W_XOUT  8192      // + e*2048   (final LN output, padded 128x16)
#define W_ATTN  16384     // + e*16384  (attn staging, 128x128)

// encoder param leaf order (dict insertion order in _enc_params)
enum { P_Wq=0,P_Wk,P_Wv,P_Wo,P_bq,P_bk,P_bv,P_bo,P_L0G,P_L0B,P_L0KG,P_L0KB,
       P_L1G,P_L1B,P_F1W,P_F1B,P_F2W,P_F2B,P_LFG,P_LFB };

struct EncArgs { const float* p[4][20]; };
struct HeadArgs { const float* h[14]; };

// ---------------- WMMA helpers (V_WMMA_F32_16X16X4_F32, chained K=16) ----------------
// A layout (16x4 f32): lane t (t<16): M=t, holds K=k0+0/k0+1; lane t+16: M=t, K=k0+2/k0+3
// B layout (4x16 f32): vgpr0 = rows {k0+0 (lanes0-15), k0+2 (lanes16-31)}, vgpr1 = {k0+1, k0+3}
// C/D layout: vgpr j: row j (lanes0-15), row j+8 (lanes16-31); col = lane&15

static __device__ inline v8f wmma_ab(const float* A, int lda, const float* B, int ldb, v8f c) {
  const int t = threadIdx.x & 31;
  const int m = t & 15;
  const int n = t & 15;
  const int koff = (t >> 4) << 1;
#pragma unroll
  for (int k0 = 0; k0 < 16; k0 += 4) {
    v2f a, b;
    a.x = A[m * lda + k0 + koff];
    a.y = A[m * lda + k0 + koff + 1];
    b.x = B[(k0 + koff) * ldb + n];
    b.y = B[(k0 + koff + 1) * ldb + n];
    c = __builtin_amdgcn_wmma_f32_16x16x4_f32(false, a, false, b, (short)0, c, false, false);
  }
  return c;
}

// B is Bsrc^T where Bsrc is NxK row-major (e.g. scores = Q @ K^T with K stored row-major)
static __device__ inline v8f wmma_abt(const float* A, int lda, const float* Bs, int ldb, v8f c) {
  const int t = threadIdx.x & 31;
  const int m = t & 15;
  const int n = t & 15;
  const int koff = (t >> 4) << 1;
#pragma unroll
  for (int k0 = 0; k0 < 16; k0 += 4) {
    v2f a, b;
    a.x = A[m * lda + k0 + koff];
    a.y = A[m * lda + k0 + koff + 1];
    b.x = Bs[n * ldb + k0 + koff];
    b.y = Bs[n * ldb + k0 + koff + 1];
    c = __builtin_amdgcn_wmma_f32_16x16x4_f32(false, a, false, b, (short)0, c, false, false);
  }
  return c;
}

static __device__ inline void store_tile(float* D, int ldd, v8f c) {
  const int t = threadIdx.x & 31;
  const int n = t & 15;
  const int rb = (t >> 4) * 8;
#pragma unroll
  for (int j = 0; j < 8; ++j) D[(rb + j) * ldd + n] = c[j];
}

static __device__ inline v8f make_c(const float* bias) {
  const int n = threadIdx.x & 15;
  float bv = bias[n];
  v8f c;
#pragma unroll
  for (int j = 0; j < 8; ++j) c[j] = bv;
  return c;
}

static __device__ inline void ln_row(const float* src, float* dst, const float* g, const float* b) {
  float v[16]; float mean = 0.f;
#pragma unroll
  for (int c = 0; c < 16; ++c) { v[c] = src[c]; mean += v[c]; }
  mean *= (1.f / 16.f);
  float var = 0.f;
#pragma unroll
  for (int c = 0; c < 16; ++c) { float d = v[c] - mean; var += d * d; }
  var *= (1.f / 16.f);
  float inv = rsqrtf(var + 1e-5f);
#pragma unroll
  for (int c = 0; c < 16; ++c) dst[c] = (v[c] - mean) * inv * g[c] + b[c];
}

// ---------------- prep: POS, eeg_t, wA, wB, wavA_before ----------------
__global__ __launch_bounds__(256) void prep_kernel(const float* __restrict__ x,
                                                   float* __restrict__ ws,
                                                   float* __restrict__ dout) {
  for (int i = threadIdx.x; i < SP * DD; i += 256) {
    int r = i >> 4, c = i & 15;
    float pv = 0.f, eg = 0.f, wa = 0.f, wb = 0.f;
    if (r < SEQ) {
      int j = c & 7;
      float f = __expf(-logf(10000.f) * (float)j / 7.f);
      float ang = (float)r * f;
      pv = (c < 8) ? sinf(ang) : cosf(ang);
      eg = x[(1 + c) * 140 + 16 + r];         // eeg_t[r][c] = x[0,0,1+c,16+r]
      wa = x[0 * 140 + c + 1 + r];            // wA[r][c] = wavA[c+1+r]
      wb = x[17 * 140 + c + 1 + r];           // wB[r][c] = wavB[c+1+r]
      dout[O_WAVB + r * 16 + c] = wa;         // wavA_before
    }
    ws[W_POS + i] = pv;
    ws[W_EEGT + i] = eg;
    ws[W_WA + i] = wa;
    ws[W_WB + i] = wb;
  }
}

// ---------------- encoder: 1 block per encoder, 4 waves, all-WMMA ----------------
__global__ __launch_bounds__(128) void enc_kernel(EncArgs ea, float* __restrict__ ws,
                                                  float* __restrict__ dout) {
  __shared__ float sX[SP][17];    // query stream (row-local evolution)
  __shared__ float sXK[SP][17];   // fixed scaled key/value input
  __shared__ float sTmp[SP][17];  // kn / qn / q / intermediate tiles
  __shared__ float sK[SP][17];
  __shared__ float sV[SP][17];
  __shared__ float sH[4][16][68]; // fc hidden staging, per wave

  const int e = blockIdx.x;
  const float* const* P = ea.p[e];
  const int tid = threadIdx.x;
  const int w = tid >> 5;
  const int lane = tid & 31;
  const int n = lane & 15;
  const int rb = (lane >> 4) * 8;

  // input selection: enc0=a2e(wA,eeg), enc1=e2a(eeg,wA), enc2=a2e2(wB,eeg), enc3=e2a2(eeg,wB)
  const float* xq = (e == 0) ? (ws + W_WA) : (e == 2) ? (ws + W_WB) : (ws + W_EEGT);
  const float* xk = (e == 1) ? (ws + W_WA) : (e == 3) ? (ws + W_WB) : (ws + W_EEGT);
  float* abuf = ws + W_ATTN + e * (SP * SP);
  float* xout = ws + W_XOUT + e * (SP * DD);
  const int attnOff = (e == 0) ? O_AEA : (e == 1) ? O_EAA : (e == 2) ? O_AEB : O_EAB;

  for (int i = tid; i < SP * DD; i += 128) {
    int r = i >> 4, c = i & 15;
    float pv = ws[W_POS + i];
    sX[r][c]  = (r < SEQ) ? 4.f * xq[i] + pv : 0.f;
    sXK[r][c] = (r < SEQ) ? 4.f * xk[i] + pv : 0.f;
  }
  __syncthreads();

  for (int l = 0; l < NL; ++l) {
    const float* Wq = P[P_Wq] + l * 256;  const float* Wk = P[P_Wk] + l * 256;
    const float* Wv = P[P_Wv] + l * 256;  const float* Wo = P[P_Wo] + l * 256;
    const float* bq = P[P_bq] + l * 16;   const float* bk = P[P_bk] + l * 16;
    const float* bv = P[P_bv] + l * 16;   const float* bo = P[P_bo] + l * 16;
    const float* l0g = P[P_L0G] + l * 16; const float* l0b = P[P_L0B] + l * 16;
    const float* l0kg = P[P_L0KG] + l * 16; const float* l0kb = P[P_L0KB] + l * 16;
    const float* l1g = P[P_L1G] + l * 16; const float* l1b = P[P_L1B] + l * 16;
    const float* f1w = P[P_F1W] + l * (16 * 64); const float* f1b = P[P_F1B] + l * 64;
    const float* f2w = P[P_F2W] + l * (64 * 16); const float* f2b = P[P_F2B] + l * 16;

    // kn for all 128 rows (per-layer LN of the fixed xk)
    ln_row(&sXK[tid][0], &sTmp[tid][0], l0kg, l0kb);
    __syncthreads();

    // K, V projections (shared across waves)
#pragma unroll
    for (int mi = 0; mi < 2; ++mi) {
      int mt = w + mi * 4;
      v8f ck = make_c(bk);
      ck = wmma_ab(&sTmp[mt * 16][0], 17, Wk, 16, ck);
      store_tile(&sK[mt * 16][0], 17, ck);
      v8f cv = make_c(bv);
      cv = wmma_ab(&sTmp[mt * 16][0], 17, Wv, 16, cv);
      store_tile(&sV[mt * 16][0], 17, cv);
    }
    __syncthreads();

    // per-wave row-tile pipeline (x evolves row-locally)
#pragma unroll
    for (int mi = 0; mi < 2; ++mi) {
      int mt = w + mi * 4;
      // qn
      if (lane < 16) ln_row(&sX[mt * 16 + lane][0], &sTmp[mt * 16 + lane][0], l0g, l0b);
      __builtin_amdgcn_wave_barrier();
      // q = qn @ Wq + bq (in-place into sTmp)
      v8f cq = make_c(bq);
      cq = wmma_ab(&sTmp[mt * 16][0], 17, Wq, 16, cq);
      store_tile(&sTmp[mt * 16][0], 17, cq);
      __builtin_amdgcn_wave_barrier();

      // scores s = q @ k^T * D^-0.5, padded cols masked
      v8f s[8];
#pragma unroll
      for (int nt = 0; nt < 8; ++nt) {
        v8f z = {};
        z = wmma_abt(&sTmp[mt * 16][0], 17, &sK[nt * 16][0], 17, z);
        s[nt] = z;
      }
#pragma unroll
      for (int nt = 0; nt < 8; ++nt) {
        int col = nt * 16 + n;
#pragma unroll
        for (int j = 0; j < 8; ++j) s[nt][j] = (col < SEQ) ? s[nt][j] * 0.25f : -3.0e38f;
      }
      // in-register row softmax (rows live on 16-lane halves)
      float mj[8], sj[8];
#pragma unroll
      for (int j = 0; j < 8; ++j) {
        float m0 = s[0][j];
#pragma unroll
        for (int nt = 1; nt < 8; ++nt) m0 = fmaxf(m0, s[nt][j]);
#pragma unroll
        for (int st = 1; st < 16; st <<= 1) m0 = fmaxf(m0, __shfl_xor(m0, st, 16));
        mj[j] = m0;
      }
#pragma unroll
      for (int j = 0; j < 8; ++j) {
        float acc = 0.f;
#pragma unroll
        for (int nt = 0; nt < 8; ++nt) { s[nt][j] = __expf(s[nt][j] - mj[j]); acc += s[nt][j]; }
#pragma unroll
        for (int st = 1; st < 16; st <<= 1) acc += __shfl_xor(acc, st, 16);
        sj[j] = 1.f / acc;
      }
#pragma unroll
      for (int nt = 0; nt < 8; ++nt)
#pragma unroll
        for (int j = 0; j < 8; ++j) s[nt][j] *= sj[j];

      // stage attn (D-layout -> memory, re-read in A-layout); also final-layer output
#pragma unroll
      for (int nt = 0; nt < 8; ++nt) {
#pragma unroll
        for (int j = 0; j < 8; ++j) {
          int grow = mt * 16 + rb + j;
          int col = nt * 16 + n;
          abuf[grow * SP + col] = s[nt][j];
          if (l == NL - 1 && grow < SEQ && col < SEQ)
            dout[attnOff + grow * SEQ + col] = s[nt][j];
        }
      }
      asm volatile("s_wait_storecnt 0x0" ::: "memory");

      // t = attn @ v
      v8f ct = {};
#pragma unroll
      for (int kt = 0; kt < 8; ++kt)
        ct = wmma_ab(&abuf[(mt * 16) * SP + kt * 16], SP, &sV[kt * 16][0], 17, ct);
      store_tile(&sTmp[mt * 16][0], 17, ct);
      __builtin_amdgcn_wave_barrier();

      // x = res + t @ Wo + bo  (residual folded into C)
      v8f co;
#pragma unroll
      for (int j = 0; j < 8; ++j) co[j] = bo[n] + sX[mt * 16 + rb + j][n];
      co = wmma_ab(&sTmp[mt * 16][0], 17, Wo, 16, co);
      store_tile(&sX[mt * 16][0], 17, co);
      __builtin_amdgcn_wave_barrier();

      // FFN: h = relu(LN(x) @ fc1_w + fc1_b)
      if (lane < 16) ln_row(&sX[mt * 16 + lane][0], &sTmp[mt * 16 + lane][0], l1g, l1b);
      __builtin_amdgcn_wave_barrier();
#pragma unroll
      for (int ht = 0; ht < 4; ++ht) {
        v8f ch = make_c(f1b + ht * 16);
        ch = wmma_ab(&sTmp[mt * 16][0], 17, f1w + ht * 16, 64, ch);
#pragma unroll
        for (int j = 0; j < 8; ++j) ch[j] = fmaxf(ch[j], 0.f);
        store_tile(&sH[w][0][ht * 16], 68, ch);
        if (l == NL - 1 && e == 0) {
#pragma unroll
          for (int j = 0; j < 8; ++j) {
            int grow = mt * 16 + rb + j;
            if (grow < SEQ) dout[O_AEFC + grow * 64 + ht * 16 + n] = ch[j];
          }
        }
      }
      __builtin_amdgcn_wave_barrier();

      // fc_o = h @ fc2_w + fc2_b ; x = res + fc_o
      v8f cf = make_c(f2b);
#pragma unroll
      for (int kt = 0; kt < 4; ++kt)
        cf = wmma_ab(&sH[w][0][kt * 16], 68, f2w + kt * 256, 16, cf);
      if (l == NL - 1 && e == 0) {
#pragma unroll
        for (int j = 0; j < 8; ++j) {
          int grow = mt * 16 + rb + j;
          if (grow < SEQ) dout[O_AEFCL + grow * 16 + n] = cf[j];
        }
      }
#pragma unroll
      for (int j = 0; j < 8; ++j) cf[j] += sX[mt * 16 + rb + j][n];
      store_tile(&sX[mt * 16][0], 17, cf);
    }
    __syncthreads();
  }

  // final LN -> workspace (and wavA_after for enc0)
  if (tid < SP) {
    float tmp[16];
    if (tid < SEQ) {
      ln_row(&sX[tid][0], tmp, P[P_LFG], P[P_LFB]);
#pragma unroll
      for (int c = 0; c < 16; ++c) {
        xout[tid * 16 + c] = tmp[c];
        if (e == 0) dout[O_WAVA + tid * 16 + c] = tmp[c];
      }
    } else {
#pragma unroll
      for (int c = 0; c < 16; ++c) xout[tid * 16 + c] = 0.f;
    }
  }
}

// ---------------- head: convs + MLPs + sigmoids (tiny, 1 block) ----------------
__global__ __launch_bounds__(256) void head_kernel(HeadArgs ha, const float* __restrict__ ws,
                                                   float* __restrict__ dout) {
  __shared__ float sD[4][SEQ][16];
  __shared__ float sT[120];   // conv-max features: [branch][conv*10+oc]
  __shared__ float sH2[120];  // fc1 hidden
  __shared__ float sF[40];    // feats concat
  __shared__ float sZ[40];    // sigmoid layer 1
  const int tid = threadIdx.x;
  const int encmap[4] = {0, 1, 3, 2};  // data=[wavA_t(a2e), eegA(e2a), eegB(e2a2), wavB_t(a2e2)]

  for (int i = tid; i < 4 * SEQ * 16; i += 256) {
    int br = i / (SEQ * 16);
    int rem = i - br * SEQ * 16;
    int r = rem >> 4, c = rem & 15;
    sD[br][r][c] = ws[W_XOUT + encmap[br] * (SP * DD) + r * 16 + c];
  }
  __syncthreads();

  if (tid < 120) {
    int i = tid / 30, k = tid % 30;
    int conv = k / 10, oc = k % 10;
    int kw = 8 + conv;
    const float* wC = ((conv == 0) ? ha.h[0] : (conv == 1) ? ha.h[2] : ha.h[4]) + (i * 10 + oc) * 16 * kw;
    const float* bC = (conv == 0) ? ha.h[1] : (conv == 1) ? ha.h[3] : ha.h[5];
    float bias = bC[i * 10 + oc];
    float best = 0.f;  // relu(..) >= 0
    for (int t0 = 0; t0 <= SEQ - kw; ++t0) {
      float acc = bias;
      for (int c = 0; c < 16; ++c)
        for (int j = 0; j < kw; ++j)
          acc += sD[i][t0 + j][c] * wC[c * kw + j];
      best = fmaxf(best, acc);
    }
    sT[tid] = best;
  }
  __syncthreads();

  if (tid < 120) {
    int i = tid / 30, j = tid % 30;
    const float* f1w = ha.h[6] + i * 900;
    float a = ha.h[7][i * 30 + j];
    for (int k = 0; k < 30; ++k) a += sT[i * 30 + k] * f1w[k * 30 + j];
    sH2[tid] = fmaxf(a, 0.f);
  }
  __syncthreads();

  if (tid < 40) {
    int i = tid / 10, o = tid % 10;
    const float* f2w = ha.h[8] + i * 300;
    float a = ha.h[9][i * 10 + o];
    for (int j = 0; j < 30; ++j) a += sH2[i * 30 + j] * f2w[j * 10 + o];
    sF[tid] = a;
  }
  __syncthreads();

  if (tid < 40) {
    float a = ha.h[11][tid];
    for (int m = 0; m < 40; ++m) a += sF[m] * ha.h[10][m * 40 + tid];
    sZ[tid] = 1.f / (1.f + __expf(-a));
  }
  __syncthreads();

  if (tid < 2) {
    float a = ha.h[13][tid];
    for (int m = 0; m < 40; ++m) a += sZ[m] * ha.h[12][m * 2 + tid];
    dout[O_OUT + tid] = 1.f / (1.f + __expf(-a));
  }
}

extern "C" void kernel_launch(void* const* d_in, const int* in_sizes, int n_in,
                              void* d_out, int out_size, void* d_ws, size_t ws_size,
                              hipStream_t stream) {
  (void)in_sizes; (void)n_in; (void)out_size; (void)ws_size;
  // d_in flattened in setup_inputs() dict insertion order:
  // [0]=x, then params: a2e(20 leaves), a2e2(20), e2a(20), e2a2(20),
  // conv1_w, conv1_b, conv2_w, conv2_b, conv3_w, conv3_b,
  // fc1_w, fc1_b, fc2_w, fc2_b, out_w1, out_b1, out_w2, out_b2
  const float* x = (const float*)d_in[0];
  float* ws = (float*)d_ws;
  float* out = (float*)d_out;

  EncArgs ea;
  // my encoder order: 0=a2e, 1=e2a, 2=a2e2, 3=e2a2
  const int base[4] = {1, 41, 21, 61};
  for (int e = 0; e < 4; ++e)
    for (int k = 0; k < 20; ++k)
      ea.p[e][k] = (const float*)d_in[base[e] + k];

  HeadArgs ha;
  for (int k = 0; k < 14; ++k) ha.h[k] = (const float*)d_in[81 + k];

  prep_kernel<<<1, 256, 0, stream>>>(x, ws, out);
  enc_kernel<<<4, 128, 0, stream>>>(ea, ws, out);
  head_kernel<<<1, 256, 0, stream>>>(ha, ws, out);
}